// SalMorphologyLearner2_83193516523626
// MI455X (gfx1250) — compile-verified
//
#include <hip/hip_runtime.h>
#include <hip/hip_bf16.h>
#include <math.h>

// ---------------------------------------------------------------------------
// SalMorphologyLearner2 for MI455X (gfx1250, wave32)
//   K1: mean-pool (bandwidth bound, coalesced)
//   K2: feat @ W1^T + b1, GELU  -- v_wmma_f32_16x16x4_f32, M=16 single tile row
//   K3: h @ W2^T + b2 (N padded 50->64) -- same WMMA
//   K4: 7x7 morphological logsumexp with LDS halo tiling
// ---------------------------------------------------------------------------

typedef __attribute__((ext_vector_type(2))) float v2f;
typedef __attribute__((ext_vector_type(8))) float v8f;

#define B_SZ   16
#define NP     1024
#define D_FEAT 256
#define IN_CH  512
#define HID    512
#define OUTC   50      // KS*KS + 1
#define OUTC_P 64      // padded for WMMA N tiles
#define IMG    256
#define KS7    7

// ---------------- K1: mean over Np for mask/patch halves --------------------
__global__ __launch_bounds__(256) void mean_kernel(const float* __restrict__ mask,
                                                   const float* __restrict__ patch,
                                                   float* __restrict__ feat) {
    const int b    = blockIdx.x;      // 0..15
    const int half = blockIdx.y;      // 0: mask, 1: patch
    const int c    = threadIdx.x;     // 0..255 (one column each; coalesced)
    const float* src = (half == 0 ? mask : patch) + (size_t)b * NP * D_FEAT + c;
    float acc = 0.0f;
    #pragma unroll 8
    for (int r = 0; r < NP; ++r) acc += src[(size_t)r * D_FEAT];
    feat[b * IN_CH + half * D_FEAT + c] = acc * (1.0f / (float)NP);
}

// ---------------- K2: h = gelu(feat @ W1^T + b1) via WMMA -------------------
// feat: (16, 512)  W1: (512, 512)  h: (16, 512). One wave per 16-col tile.
__global__ __launch_bounds__(32) void gemm1_kernel(const float* __restrict__ feat,
                                                   const float* __restrict__ W1,
                                                   const float* __restrict__ b1,
                                                   float* __restrict__ h) {
    const int tile = blockIdx.x;          // 0..31 -> n0 = tile*16
    const int lane = threadIdx.x;         // 0..31
    const int hi   = lane >> 4;           // 0: K pair {k,k+1}, 1: {k+2,k+3}
    const int l    = lane & 15;           // M row (A) / N col (B,C)
    const int n    = tile * 16 + l;

    v8f acc = {};
    for (int k = 0; k < IN_CH; k += 4) {
        const int ka = k + hi * 2;
        v2f a, bfrag;
        // A 16x4 f32: lanes0-15 VGPR0=K0,VGPR1=K1 ; lanes16-31 K2,K3 (same M rows)
        a.x = feat[l * IN_CH + ka];
        a.y = feat[l * IN_CH + ka + 1];
        // B 4x16 f32: B[k][n] = W1[n][k]; low lanes K={k,k+1}, high lanes K={k+2,k+3}
        bfrag.x = W1[(size_t)n * IN_CH + ka];
        bfrag.y = W1[(size_t)n * IN_CH + ka + 1];
        acc = __builtin_amdgcn_wmma_f32_16x16x4_f32(false, a, false, bfrag,
                                                    (short)0, acc, false, false);
    }
    // C layout: VGPR r -> M = r + 8*hi, N = l (+ tile base); exact GELU
    const float bias = b1[n];
    #pragma unroll
    for (int r = 0; r < 8; ++r) {
        const int row = r + hi * 8;
        float x = acc[r] + bias;
        float g = 0.5f * x * (1.0f + erff(x * 0.70710678118654752f));
        h[row * HID + n] = g;
    }
}

// ---------------- K3: params = h @ W2^T + b2 via WMMA (N padded) ------------
// h: (16, 512)  W2: (50, 512)  params: (16, 64) with cols >= 50 ignored.
__global__ __launch_bounds__(128) void gemm2_kernel(const float* __restrict__ h,
                                                    const float* __restrict__ W2,
                                                    const float* __restrict__ b2,
                                                    float* __restrict__ params) {
    const int wave = threadIdx.x >> 5;    // 0..3 -> n0 = wave*16
    const int lane = threadIdx.x & 31;
    const int hi   = lane >> 4;
    const int l    = lane & 15;
    const int n    = wave * 16 + l;       // 0..63
    const bool nv  = (n < OUTC);

    v8f acc = {};
    for (int k = 0; k < HID; k += 4) {
        const int ka = k + hi * 2;
        v2f a, bfrag;
        a.x = h[l * HID + ka];
        a.y = h[l * HID + ka + 1];
        bfrag.x = nv ? W2[(size_t)n * HID + ka]     : 0.0f;
        bfrag.y = nv ? W2[(size_t)n * HID + ka + 1] : 0.0f;
        acc = __builtin_amdgcn_wmma_f32_16x16x4_f32(false, a, false, bfrag,
                                                    (short)0, acc, false, false);
    }
    const float bias = nv ? b2[n] : 0.0f;
    #pragma unroll
    for (int r = 0; r < 8; ++r) {
        const int row = r + hi * 8;
        params[row * OUTC_P + n] = acc[r] + bias;
    }
}

// ---------------- K4: morphological soft-logsumexp 7x7 ----------------------
// One block = one (batch, output row). LDS: halo rows + symmetrized kernel.
__global__ __launch_bounds__(256) void morpho_kernel(const float* __restrict__ x,
                                                     const float* __restrict__ params,
                                                     float* __restrict__ out) {
    const int b    = blockIdx.y;
    const int hrow = blockIdx.x;          // 0..255
    const int c    = threadIdx.x;         // 0..255 (output column)

    __shared__ float w_sh[KS7 * KS7];
    __shared__ float s_sh;
    __shared__ float x_sh[KS7][IMG + 8];  // cols -3..258 stored at [cc] = gc+3

    const float* p = params + b * OUTC_P;
    if (c < KS7 * KS7) {
        const int i = c / KS7, j = c % KS7;
        const float w1v = 1.0f / (1.0f + __expf(-p[c]));
        const float w2v = 1.0f / (1.0f + __expf(-p[(KS7 - 1 - i) * KS7 + (KS7 - 1 - j)]));
        w_sh[c] = 0.5f * (w1v + w2v);
    }
    if (c == 64) {
        const float t = tanhf(p[KS7 * KS7]);           // T_SIGN = 1
        s_sh = (fabsf(t) >= 1e-7f) ? t : 1.0f;         // m*sign + (1-m)
    }
    __syncthreads();

    const float s = s_sh;
    const float* xb = x + (size_t)b * IMG * IMG;

    // Stage rows hrow-3 .. hrow+3, cols -3 .. 258 of s*x (OOB -> 0, and the
    // zeros DO participate in the logsumexp, matching jnp.pad semantics).
    for (int idx = c; idx < KS7 * (IMG + 6); idx += 256) {
        const int r  = idx / (IMG + 6);
        const int cc = idx % (IMG + 6);
        const int gr = hrow - 3 + r;
        const int gc = cc - 3;
        float v = 0.0f;
        if (gr >= 0 && gr < IMG && gc >= 0 && gc < IMG)
            v = s * xb[gr * IMG + gc];
        x_sh[r][cc] = v;
    }
    __syncthreads();

    // Online (streaming) stable logsumexp over 49 taps; ALPHA == 1.
    float m = -3.402823466e38f;
    float sum = 0.0f;
    #pragma unroll
    for (int i = 0; i < KS7; ++i) {
        #pragma unroll
        for (int j = 0; j < KS7; ++j) {
            const float v = x_sh[i][c + j] + w_sh[i * KS7 + j];
            if (v > m) { sum = sum * __expf(m - v) + 1.0f; m = v; }
            else       { sum += __expf(v - m); }
        }
    }
    out[((size_t)b * IMG + hrow) * IMG + c] = s * (m + __logf(sum));
}

// ---------------------------------------------------------------------------
extern "C" void kernel_launch(void* const* d_in, const int* in_sizes, int n_in,
                              void* d_out, int out_size, void* d_ws, size_t ws_size,
                              hipStream_t stream) {
    const float* mask_feats    = (const float*)d_in[0];   // (16,1024,256)
    const float* patch_feats   = (const float*)d_in[1];   // (16,1024,256)
    const float* low_res_masks = (const float*)d_in[2];   // (16,1,256,256)
    const float* W1            = (const float*)d_in[3];   // (512,512)
    const float* b1            = (const float*)d_in[4];   // (512,)
    const float* W2            = (const float*)d_in[5];   // (50,512)
    const float* b2            = (const float*)d_in[6];   // (50,)
    float*       out           = (float*)d_out;           // (16,1,256,256)

    float* ws     = (float*)d_ws;
    float* feat   = ws;                    // 16*512
    float* h      = ws + B_SZ * IN_CH;     // 16*512
    float* params = h  + B_SZ * HID;       // 16*64

    mean_kernel<<<dim3(B_SZ, 2), 256, 0, stream>>>(mask_feats, patch_feats, feat);
    gemm1_kernel<<<HID / 16, 32, 0, stream>>>(feat, W1, b1, h);
    gemm2_kernel<<<1, 128, 0, stream>>>(h, W2, b2, params);
    morpho_kernel<<<dim3(IMG, B_SZ), 256, 0, stream>>>(low_res_masks, params, out);
}